// GCN_43173011259684
// MI455X (gfx1250) — compile-verified
//
#include <hip/hip_runtime.h>
#include <math.h>

typedef float v2f __attribute__((ext_vector_type(2)));
typedef float v8f __attribute__((ext_vector_type(8)));
typedef int   v4i __attribute__((ext_vector_type(4)));

#if __has_builtin(__builtin_amdgcn_global_load_async_to_lds_b128) && \
    __has_builtin(__builtin_amdgcn_s_wait_asynccnt)
#define GCN_USE_ASYNC_LDS 1
typedef __attribute__((address_space(1))) v4i* gcn_gptr_t;
typedef __attribute__((address_space(3))) v4i* gcn_lptr_t;
#else
#define GCN_USE_ASYNC_LDS 0
#endif

// ---------------------------------------------------------------------------
// Degree / normalization kernels
// ---------------------------------------------------------------------------
__global__ void gcn_deg_init(float* deg, int n) {
    int i = blockIdx.x * blockDim.x + threadIdx.x;
    if (i < n) deg[i] = 1.0f;                       // self-loop contributes 1
}

__global__ void gcn_deg_count(const int* __restrict__ dst, float* deg, int e) {
    int i = blockIdx.x * blockDim.x + threadIdx.x;
    if (i < e) atomicAdd(&deg[dst[i]], 1.0f);
}

__global__ void gcn_deg_rsqrt(float* deg, int n) {
    int i = blockIdx.x * blockDim.x + threadIdx.x;
    if (i < n) deg[i] = rsqrtf(deg[i]);             // deg >= 1 always (self loops)
}

// ---------------------------------------------------------------------------
// FP32 WMMA GEMM:  H[N x OUT] = A[N x K] * W[K x OUT]   (row-major)
// Block = 256 threads = 8 waves. Block covers 256 rows x one 64-col group.
// W tile [K x 64] (64 KB) staged in LDS once per block (async-to-LDS if
// available). Each wave computes a 32x64 tile: 8 accumulators, 8 WMMAs per
// K-step against 2 A-fragment loads + LDS B reads.
// ---------------------------------------------------------------------------
template<int K, int OUT>
__global__ __launch_bounds__(256)
void gcn_gemm_wmma_f32(const float* __restrict__ A, const float* __restrict__ W,
                       float* __restrict__ H, int N) {
    __shared__ float wlds[K * 64];                  // 64 KB for K=256

    const int tid  = threadIdx.x;
    const int lane = tid & 31;                      // wave32
    const int wave = tid >> 5;                      // 0..7
    const int half = lane >> 4;                     // 0/1
    const int l16  = lane & 15;

    const int n0 = blockIdx.y * 64;

    // ---- stage W[K x 64] tile into LDS (16B chunks, coalesced) ----
#if GCN_USE_ASYNC_LDS
    #pragma unroll
    for (int c = tid; c < (K * 64) / 4; c += 256) {
        const int k   = c >> 4;                     // (c*4)/64
        const int col = (c & 15) << 2;              // (c*4)%64
        __builtin_amdgcn_global_load_async_to_lds_b128(
            (gcn_gptr_t)(W + (size_t)k * OUT + n0 + col),
            (gcn_lptr_t)(wlds + c * 4), 0, 0);
    }
    __builtin_amdgcn_s_wait_asynccnt(0);
    __syncthreads();
#else
    #pragma unroll
    for (int c = tid; c < (K * 64) / 4; c += 256) {
        const int k   = c >> 4;
        const int col = (c & 15) << 2;
        *(float4*)(wlds + c * 4) = *(const float4*)(W + (size_t)k * OUT + n0 + col);
    }
    __syncthreads();
#endif

    // ---- per-wave 32x64 tile ----
    const int m0 = (blockIdx.x * 8 + wave) * 32;

    int arow0 = m0 + l16;      if (arow0 >= N) arow0 = N - 1;   // clamp tails
    int arow1 = m0 + 16 + l16; if (arow1 >= N) arow1 = N - 1;
    const float* ap0 = A + (size_t)arow0 * K;
    const float* ap1 = A + (size_t)arow1 * K;

    v8f acc00 = {}, acc01 = {}, acc02 = {}, acc03 = {};
    v8f acc10 = {}, acc11 = {}, acc12 = {}, acc13 = {};

    #pragma unroll 2
    for (int k0 = 0; k0 < K; k0 += 4) {
        const int ka = k0 + 2 * half;
        // A fragments: V0 = A[m][ka], V1 = A[m][ka+1]  (contiguous 8B load)
        v2f a0 = *(const v2f*)(ap0 + ka);
        v2f a1 = *(const v2f*)(ap1 + ka);

        // B fragments from LDS: V0 = W[ka][n], V1 = W[ka+1][n]
        const float* wr0 = wlds + ka * 64 + l16;
        const float* wr1 = wr0 + 64;
        v2f b0 = { wr0[0],  wr1[0]  };
        v2f b1 = { wr0[16], wr1[16] };
        v2f b2 = { wr0[32], wr1[32] };
        v2f b3 = { wr0[48], wr1[48] };

        acc00 = __builtin_amdgcn_wmma_f32_16x16x4_f32(false, a0, false, b0, (short)0, acc00, false, false);
        acc01 = __builtin_amdgcn_wmma_f32_16x16x4_f32(false, a0, false, b1, (short)0, acc01, false, false);
        acc02 = __builtin_amdgcn_wmma_f32_16x16x4_f32(false, a0, false, b2, (short)0, acc02, false, false);
        acc03 = __builtin_amdgcn_wmma_f32_16x16x4_f32(false, a0, false, b3, (short)0, acc03, false, false);
        acc10 = __builtin_amdgcn_wmma_f32_16x16x4_f32(false, a1, false, b0, (short)0, acc10, false, false);
        acc11 = __builtin_amdgcn_wmma_f32_16x16x4_f32(false, a1, false, b1, (short)0, acc11, false, false);
        acc12 = __builtin_amdgcn_wmma_f32_16x16x4_f32(false, a1, false, b2, (short)0, acc12, false, false);
        acc13 = __builtin_amdgcn_wmma_f32_16x16x4_f32(false, a1, false, b3, (short)0, acc13, false, false);
    }

    // C layout: VGPR r -> row (r + 8*half), col = l16 (+16/+32/+48)
    #pragma unroll
    for (int r = 0; r < 8; ++r) {
        const int row0 = m0 + r + 8 * half;
        if (row0 < N) {
            float* out = H + (size_t)row0 * OUT + n0 + l16;
            out[0] = acc00[r]; out[16] = acc01[r]; out[32] = acc02[r]; out[48] = acc03[r];
        }
        const int row1 = m0 + 16 + r + 8 * half;
        if (row1 < N) {
            float* out = H + (size_t)row1 * OUT + n0 + l16;
            out[0] = acc10[r]; out[16] = acc11[r]; out[32] = acc12[r]; out[48] = acc13[r];
        }
    }
}

// ---------------------------------------------------------------------------
// Aggregation
// ---------------------------------------------------------------------------
// Self-loop term: agg[i,:] = H[i,:] * dis[i]^2
template<int OUT>
__global__ void gcn_init_agg(const float* __restrict__ H, const float* __restrict__ dis,
                             float* __restrict__ agg, int N) {
    size_t idx = (size_t)blockIdx.x * blockDim.x + threadIdx.x;
    size_t tot = (size_t)N * OUT;
    if (idx >= tot) return;
    int i = (int)(idx / OUT);
    float d = dis[i];
    agg[idx] = H[idx] * d * d;
}

// Edge term: agg[dst,:] += H[src,:] * dis[src]*dis[dst]  (float4 per thread)
template<int OUT>
__global__ void gcn_edge_scatter(const float* __restrict__ H,
                                 const int* __restrict__ src, const int* __restrict__ dst,
                                 const float* __restrict__ dis,
                                 float* __restrict__ agg, int E) {
    constexpr int CHUNKS = OUT >> 2;
    long long idx = (long long)blockIdx.x * blockDim.x + threadIdx.x;
    long long tot = (long long)E * CHUNKS;
    if (idx >= tot) return;
    int e = (int)(idx / CHUNKS);
    int c = (int)(idx % CHUNKS) * 4;
    int s = src[e], d = dst[e];
    float w = dis[s] * dis[d];
    const float4 h = *(const float4*)(H + (size_t)s * OUT + c);
    float* ap = agg + (size_t)d * OUT + c;
    atomicAdd(ap + 0, h.x * w);
    atomicAdd(ap + 1, h.y * w);
    atomicAdd(ap + 2, h.z * w);
    atomicAdd(ap + 3, h.w * w);
}

template<int OUT>
__global__ void gcn_bias_act(float* __restrict__ X, const float* __restrict__ b,
                             int N, int relu) {
    size_t idx = (size_t)blockIdx.x * blockDim.x + threadIdx.x;
    size_t tot = (size_t)N * OUT;
    if (idx >= tot) return;
    int f = (int)(idx % OUT);
    float v = X[idx] + b[f];
    if (relu) v = fmaxf(v, 0.0f);
    X[idx] = v;
}

// ---------------------------------------------------------------------------
// Pool + log-softmax
// ---------------------------------------------------------------------------
__global__ void gcn_zero(float* p, int n) {
    int i = blockIdx.x * blockDim.x + threadIdx.x;
    if (i < n) p[i] = 0.0f;
}

// blockDim.x == C; each block reduces 256 rows, one atomic per thread
__global__ void gcn_pool_sum(const float* __restrict__ X, float* __restrict__ pooled,
                             int N, int C) {
    int c = threadIdx.x;
    int r0 = blockIdx.x * 256;
    int r1 = min(r0 + 256, N);
    float s = 0.0f;
    for (int r = r0; r < r1; ++r) s += X[(size_t)r * C + c];
    atomicAdd(&pooled[c], s);
}

// single block, blockDim.x == C (=128): out[0:C]=pooled, out[C:2C]=log_softmax
__global__ void gcn_logsoftmax(const float* __restrict__ pooled, float* __restrict__ out, int C) {
    __shared__ float red[256];
    int c = threadIdx.x;
    float v = pooled[c];
    red[c] = v;
    __syncthreads();
    for (int s = C >> 1; s > 0; s >>= 1) {
        if (c < s) red[c] = fmaxf(red[c], red[c + s]);
        __syncthreads();
    }
    float mx = red[0];
    __syncthreads();
    red[c] = expf(v - mx);
    __syncthreads();
    for (int s = C >> 1; s > 0; s >>= 1) {
        if (c < s) red[c] += red[c + s];
        __syncthreads();
    }
    float lse = logf(red[0]) + mx;
    out[c] = v;
    out[C + c] = v - lse;
}

// ---------------------------------------------------------------------------
// Launch
// ---------------------------------------------------------------------------
extern "C" void kernel_launch(void* const* d_in, const int* in_sizes, int n_in,
                              void* d_out, int out_size, void* d_ws, size_t ws_size,
                              hipStream_t stream) {
    const float* x  = (const float*)d_in[0];
    const int*   ei = (const int*)d_in[1];
    const float* W1 = (const float*)d_in[2]; const float* b1 = (const float*)d_in[3];
    const float* W2 = (const float*)d_in[4]; const float* b2 = (const float*)d_in[5];
    const float* W3 = (const float*)d_in[6]; const float* b3 = (const float*)d_in[7];
    const float* W4 = (const float*)d_in[8]; const float* b4 = (const float*)d_in[9];

    constexpr int FIN = 256, HID = 256, CLS = 128;
    const int N = in_sizes[0] / FIN;         // 50000
    const int E = in_sizes[1] / 2;           // 300000

    const int* src = ei;
    const int* dst = ei + E;

    float* hbuf   = (float*)d_ws;                     // GEMM output      [N x 256]
    float* abuf   = hbuf + (size_t)N * HID;           // aggregate/input  [N x 256]
    float* dis    = abuf + (size_t)N * HID;           // [N]
    float* pooled = dis + N;                          // [CLS]

    // --- normalization: deg (with self loop) -> rsqrt ---
    gcn_deg_init <<<(N + 255) / 256, 256, 0, stream>>>(dis, N);
    gcn_deg_count<<<(E + 255) / 256, 256, 0, stream>>>(dst, dis, E);
    gcn_deg_rsqrt<<<(N + 255) / 256, 256, 0, stream>>>(dis, N);

    auto layer_hid = [&](const float* in, const float* W, const float* b) {
        dim3 g((N + 255) / 256, HID / 64);
        gcn_gemm_wmma_f32<256, HID><<<g, 256, 0, stream>>>(in, W, hbuf, N);
        size_t tot = (size_t)N * HID;
        unsigned nb = (unsigned)((tot + 255) / 256);
        gcn_init_agg<HID><<<nb, 256, 0, stream>>>(hbuf, dis, abuf, N);
        long long etot = (long long)E * (HID >> 2);
        unsigned eb = (unsigned)((etot + 255) / 256);
        gcn_edge_scatter<HID><<<eb, 256, 0, stream>>>(hbuf, src, dst, dis, abuf, E);
        gcn_bias_act<HID><<<nb, 256, 0, stream>>>(abuf, b, N, 1);
    };

    layer_hid(x,    W1, b1);
    layer_hid(abuf, W2, b2);
    layer_hid(abuf, W3, b3);

    {   // layer 4: OUT = CLS = 128, no relu
        dim3 g((N + 255) / 256, CLS / 64);
        gcn_gemm_wmma_f32<256, CLS><<<g, 256, 0, stream>>>(abuf, W4, hbuf, N);
        size_t tot = (size_t)N * CLS;
        unsigned nb = (unsigned)((tot + 255) / 256);
        gcn_init_agg<CLS><<<nb, 256, 0, stream>>>(hbuf, dis, abuf, N);
        long long etot = (long long)E * (CLS >> 2);
        unsigned eb = (unsigned)((etot + 255) / 256);
        gcn_edge_scatter<CLS><<<eb, 256, 0, stream>>>(hbuf, src, dst, dis, abuf, E);
        gcn_bias_act<CLS><<<nb, 256, 0, stream>>>(abuf, b4, N, 0);
    }

    gcn_zero      <<<1, CLS, 0, stream>>>(pooled, CLS);
    gcn_pool_sum  <<<(N + 255) / 256, CLS, 0, stream>>>(abuf, pooled, N, CLS);
    gcn_logsoftmax<<<1, CLS, 0, stream>>>(pooled, (float*)d_out, CLS);
}